// RandomSampleNeighbourPts_29248727286340
// MI455X (gfx1250) — compile-verified
//
#include <hip/hip_runtime.h>
#include <math.h>

// Problem constants from the reference
#define H_     352
#define W_     1216
#define B_     10
#define PTS_   5000
#define DENSE_ 20

typedef float v2f __attribute__((ext_vector_type(2)));
typedef float v8f __attribute__((ext_vector_type(8)));

// ---------------------------------------------------------------------------
// One wave handles 16 sample points.
//   Phase 1: gather 20 (mask, disp) neighbor pairs per point, reduce the five
//            per-point statistics  St=Σt, Sd=Σd, Sdt=Σdt, Sd2=Σd², Sd2t=Σd²t
//            with chained V_WMMA_F32_16X16X4_F32 against a ones B-matrix.
//            (A is 16 points × 4 neighbors per chunk; 5 chunks cover K=20.)
//   Phase 2: lanes {0..7, 16..23} each own one point (matching the C/D VGPR
//            layout), evaluate the 5x5-patch sobel/maxpool sel1 test and the
//            mean/variance math, then a deterministic wave+block reduction.
// ---------------------------------------------------------------------------
__global__ void __launch_bounds__(256)
point_stats_kernel(const float* __restrict__ disp,
                   const float* __restrict__ fmask,
                   const int*   __restrict__ cx,
                   const int*   __restrict__ cy,
                   const int*   __restrict__ bx,
                   const int*   __restrict__ by,
                   float*       __restrict__ blockPartials,
                   int npts)
{
    __shared__ float smem[8][5];

    const int lane = threadIdx.x & 31;
    const int wave = threadIdx.x >> 5;
    const int waveGlobal = blockIdx.x * 8 + wave;
    const int base = waveGlobal * 16;          // first point of this wave

    float r0 = 0.0f, r1 = 0.0f, r2 = 0.0f, r3 = 0.0f, r4 = 0.0f;

    if (base < npts) {                         // wave-uniform: EXEC stays all-ones
        // ---------------- Phase 1: WMMA neighbor statistics ----------------
        // A-matrix 16x4 layout: lane L owns row M=L&15; lanes<16 hold K=0,1,
        // lanes>=16 hold K=2,3 (VGPR0/VGPR1).
        const int m  = lane & 15;
        int gp = base + m; if (gp >= npts) gp = npts - 1;   // safe clamp (npts%16==0 normally)
        const int b   = gp / PTS_;
        const int pcx = cx[gp];
        const int pcy = cy[gp];
        const int cb  = b * (H_ * W_);
        const int kb  = (lane < 16) ? 0 : 2;

        v8f Ct = {}, Cd = {}, Cdt = {}, Cd2 = {}, Cd2t = {};
        const v2f Bones = {1.0f, 1.0f};        // D = A x ones  ==> row sums

        #pragma unroll
        for (int c = 0; c < 5; ++c) {          // 5 K-chunks of 4 neighbors
            v2f At, Ad, Adt, Ad2, Ad2t;
            #pragma unroll
            for (int u = 0; u < 2; ++u) {
                const int j  = c * 4 + kb + u;            // neighbor index 0..19
                const int sx = pcx + bx[gp * DENSE_ + j];
                const int sy = pcy + by[gp * DENSE_ + j];
                const int idx = cb + sy * W_ + sx;
                const float t = fmask[idx];
                const float d = disp[idx];
                At[u]   = t;
                Ad[u]   = d;
                Adt[u]  = d * t;
                Ad2[u]  = d * d;
                Ad2t[u] = d * d * t;
            }
            Ct   = __builtin_amdgcn_wmma_f32_16x16x4_f32(false, At,   false, Bones, (short)0, Ct,   false, false);
            Cd   = __builtin_amdgcn_wmma_f32_16x16x4_f32(false, Ad,   false, Bones, (short)0, Cd,   false, false);
            Cdt  = __builtin_amdgcn_wmma_f32_16x16x4_f32(false, Adt,  false, Bones, (short)0, Cdt,  false, false);
            Cd2  = __builtin_amdgcn_wmma_f32_16x16x4_f32(false, Ad2,  false, Bones, (short)0, Cd2,  false, false);
            Cd2t = __builtin_amdgcn_wmma_f32_16x16x4_f32(false, Ad2t, false, Bones, (short)0, Cd2t, false, false);
        }

        // ------------- Phase 2: per-point post-processing ------------------
        // C/D 16x16 layout: VGPR rr holds row M=rr (lanes 0-15) / M=rr+8
        // (lanes 16-31); all 16 columns are identical (B was ones).
        // Lane L with (L&8)==0 owns point q = (L&7) + 8*(L>=16) in C[L&7].
        if ((lane & 8) == 0) {
            const int q  = (lane & 7) | ((lane & 16) >> 1);
            const int gq = base + q;
            if (gq < npts) {
                float st = 0.f, sd = 0.f, sdt = 0.f, sd2 = 0.f, sd2t = 0.f;
                #pragma unroll
                for (int rr = 0; rr < 8; ++rr) {
                    if ((lane & 7) == rr) {
                        st = Ct[rr]; sd = Cd[rr]; sdt = Cdt[rr];
                        sd2 = Cd2[rr]; sd2t = Cd2t[rr];
                    }
                }

                // sel1: |sobel_x(mask)| maxpooled over 3x3, gated by disp.
                // Sample points are >=11 from every edge, so the 5x5 patch
                // never touches padding or the zeroed 4-pixel border.
                const int bq  = gq / PTS_;
                const int qcx = cx[gq];
                const int qcy = cy[gq];
                const int cbq = bq * (H_ * W_);
                const float* mp = fmask + cbq + (qcy - 2) * W_ + (qcx - 2);
                float P[5][5];
                #pragma unroll
                for (int yy = 0; yy < 5; ++yy)
                    #pragma unroll
                    for (int xx = 0; xx < 5; ++xx)
                        P[yy][xx] = mp[yy * W_ + xx];
                float mx = 0.0f;
                #pragma unroll
                for (int yy = 1; yy <= 3; ++yy)
                    #pragma unroll
                    for (int xx = 1; xx <= 3; ++xx) {
                        const float s = (P[yy-1][xx+1] - P[yy-1][xx-1])
                                + 2.0f * (P[yy  ][xx+1] - P[yy  ][xx-1])
                                +        (P[yy+1][xx+1] - P[yy+1][xx-1]);
                        mx = fmaxf(mx, fabsf(s));
                    }
                const float dc = disp[cbq + qcy * W_ + qcx];
                // mg = max(mx*ind, 3) - 3 > 0.1  <=>  ind && mx > 3.1
                const bool sel1 = (dc > 0.007f) && (mx > 3.1f);

                const float posNum = st;
                const float negNum = (float)DENSE_ - st;   // mask is exactly 0/1
                const bool balance = (posNum > 4.0f) && (negNum > 4.0f);
                const bool valid   = sel1 && balance;
                const float w = valid ? 1.0f : 0.0f;

                const float posNumS = fmaxf(posNum, 1.0f);
                const float negNumS = fmaxf(negNum, 1.0f);
                const float posMean = sdt / posNumS;
                const float negSumD = sd - sdt;
                const float negMean = negSumD / negNumS;
                // Sum (d - mean)^2 * sel  via expanded form
                const float varPosN = sd2t - 2.0f * posMean * sdt
                                    + posMean * posMean * st;
                const float varNegN = (sd2 - sd2t) - 2.0f * negMean * negSumD
                                    + negMean * negMean * negNum;
                const float simPos = sqrtf(fmaxf(varPosN, 0.0f) / posNumS + 1e-14f);
                const float simNeg = sqrtf(fmaxf(varNegN, 0.0f) / negNumS + 1e-14f);

                r0 = sel1 ? 1.0f : 0.0f;
                r1 = w;
                r2 = simPos * w;
                r3 = simNeg * w;
                r4 = (negMean - posMean) * w;
            }
        }
    }

    // ------------- deterministic wave32 + block reduction ------------------
    float rv[5] = { r0, r1, r2, r3, r4 };
    #pragma unroll
    for (int s = 0; s < 5; ++s) {
        float v = rv[s];
        #pragma unroll
        for (int off = 16; off > 0; off >>= 1)
            v += __shfl_xor(v, off, 32);
        if (lane == 0) smem[wave][s] = v;
    }
    __syncthreads();
    if (threadIdx.x == 0) {
        float acc[5] = {0.f, 0.f, 0.f, 0.f, 0.f};
        for (int wv = 0; wv < 8; ++wv)
            for (int s = 0; s < 5; ++s)
                acc[s] += smem[wv][s];
        #pragma unroll
        for (int s = 0; s < 5; ++s)
            blockPartials[blockIdx.x * 8 + s] = acc[s];
        blockPartials[blockIdx.x * 8 + 5] = 0.0f;
        blockPartials[blockIdx.x * 8 + 6] = 0.0f;
        blockPartials[blockIdx.x * 8 + 7] = 0.0f;
    }
}

// Single-thread, fixed-order finalize: deterministic across graph replays.
__global__ void finalize_kernel(const float* __restrict__ blockPartials,
                                int nBlocks,
                                float* __restrict__ out)
{
    if (threadIdx.x == 0 && blockIdx.x == 0) {
        float a0 = 0.f, a1 = 0.f, a2 = 0.f, a3 = 0.f, a4 = 0.f;
        for (int i = 0; i < nBlocks; ++i) {
            a0 += blockPartials[i * 8 + 0];
            a1 += blockPartials[i * 8 + 1];
            a2 += blockPartials[i * 8 + 2];
            a3 += blockPartials[i * 8 + 3];
            a4 += blockPartials[i * 8 + 4];
        }
        const float nvalid  = a1;
        const float nvalidS = fmaxf(nvalid, 1.0f);
        const float lossSim = (a2 / nvalidS + a3 / nvalidS) * 0.5f;
        const float lossContrast = a4 / nvalidS + 0.02f;
        const bool ok = (a0 >= 100.0f) && (nvalid >= 100.0f);
        out[0] = ok ? lossSim : 0.0f;
        out[1] = ok ? lossContrast : 0.0f;
    }
}

extern "C" void kernel_launch(void* const* d_in, const int* in_sizes, int n_in,
                              void* d_out, int out_size, void* d_ws, size_t ws_size,
                              hipStream_t stream)
{
    const float* disp  = (const float*)d_in[0];
    const float* fmask = (const float*)d_in[1];
    const int*   cx    = (const int*)d_in[2];
    const int*   cy    = (const int*)d_in[3];
    const int*   bx    = (const int*)d_in[4];
    const int*   by    = (const int*)d_in[5];
    float* out = (float*)d_out;

    const int npts        = in_sizes[2];               // B*PTS = 50000
    const int wavesNeeded = (npts + 15) / 16;          // 16 points per wave
    const int nBlocks     = (wavesNeeded + 7) / 8;     // 8 waves (256 thr) per block

    float* partials = (float*)d_ws;                    // nBlocks * 8 floats

    point_stats_kernel<<<nBlocks, 256, 0, stream>>>(disp, fmask, cx, cy, bx, by,
                                                    partials, npts);
    finalize_kernel<<<1, 32, 0, stream>>>(partials, nBlocks, out);
}